// RelationGraphConvolution_withBasisRegularization_16982300688783
// MI455X (gfx1250) — compile-verified
//
#include <hip/hip_runtime.h>

// R-GCN with basis regularization for MI455X (gfx1250, wave32).
// out = relu( Σ_r  A_r · (x @ (Σ_b comp[r,b]·W_basis[b])) )
//
// Phase cost model: GEMM = 13.1 GFLOP fp32 (WMMA f32 16x16x4);
// scatter = ~4.2 GB random HBM traffic (bandwidth bound at 23.3 TB/s).

typedef __attribute__((ext_vector_type(2))) float v2f;
typedef __attribute__((ext_vector_type(8))) float v8f;

#define F       128   // F_IN == F_OUT
#define NREL    8
#define NBASIS  4

// ---------------- utility kernels ----------------

__global__ __launch_bounds__(256) void zero_f32(float* __restrict__ p, long long n) {
    long long i = (long long)blockIdx.x * blockDim.x + threadIdx.x;
    long long stride = (long long)gridDim.x * blockDim.x;
    for (; i < n; i += stride) p[i] = 0.0f;
}

__global__ __launch_bounds__(256) void relu_inplace(float* __restrict__ p, long long n) {
    long long i = (long long)blockIdx.x * blockDim.x + threadIdx.x;
    long long stride = (long long)gridDim.x * blockDim.x;
    for (; i < n; i += stride) p[i] = fmaxf(p[i], 0.0f);
}

// W_rel[r,i,o] = sum_b W_comp[r,b] * W_basis[b,i,o]   (8*128*128 outputs)
__global__ __launch_bounds__(256) void wrel_kernel(const float* __restrict__ Wb,
                                                   const float* __restrict__ Wc,
                                                   float* __restrict__ Wr) {
    int idx = blockIdx.x * 256 + threadIdx.x;   // [0, NREL*F*F)
    int r  = idx >> 14;                         // / (128*128)
    int io = idx & (F * F - 1);
    float acc = 0.0f;
#pragma unroll
    for (int b = 0; b < NBASIS; ++b)
        acc = fmaf(Wc[r * NBASIS + b], Wb[b * F * F + io], acc);
    Wr[idx] = acc;
}

// ---------------- fp32 WMMA GEMM: pre = x @ Wr  (M x 128 x 128) ----------------
// Block: 256 threads = 8 waves. Block tile: 16 rows x 128 cols.
// Wave w computes the 16x16 tile at columns [16w, 16w+16).
// A (16x4 f32) lane layout: lanes 0-15 hold M=lane, K={k,k+1} in {V0,V1};
//                           lanes 16-31 hold M=lane-16, K={k+2,k+3}.
// B (4x16 f32) mirrors:     lanes 0-15 hold N=lane, K={k,k+1};
//                           lanes 16-31 hold N=lane-16, K={k+2,k+3}.
// C/D (16x16 f32, 8 VGPRs): lane col = lane%16, VGPR v holds row v + 8*(lane>=16).
__global__ __launch_bounds__(256) void gemm_wmma(const float* __restrict__ x,
                                                 const float* __restrict__ Wr,
                                                 float* __restrict__ pre,
                                                 int n_nodes) {
    __shared__ float sX[16 * (F + 1)];          // +1 pad: conflict-free column reads
    const int tid  = threadIdx.x;
    const int row0 = blockIdx.x * 16;

    // Stage 16 contiguous x-rows (8 KB) into LDS, fully coalesced.
    for (int i = tid; i < 16 * F; i += 256)
        sX[(i >> 7) * (F + 1) + (i & (F - 1))] = x[(size_t)row0 * F + i];
    __syncthreads();

    const int wave = tid >> 5;
    const int lane = tid & 31;
    const int hi   = lane >> 4;       // half-wave select
    const int l16  = lane & 15;
    const int col  = wave * 16 + l16; // output column owned by this lane

    const float* sxrow = sX + l16 * (F + 1);
    v8f c = {};                       // f32 accumulator (8 VGPRs)

#pragma unroll 4
    for (int k = 0; k < F; k += 4) {
        const int ka = k + (hi << 1); // lanes 16-31 take K+2, K+3
        v2f a, b;
        a.x = sxrow[ka];
        a.y = sxrow[ka + 1];
        b.x = Wr[ka * F + col];       // W_rel is 64 KB: L2/WGP$ resident
        b.y = Wr[(ka + 1) * F + col];
        c = __builtin_amdgcn_wmma_f32_16x16x4_f32(
                /*neg_a=*/false, a, /*neg_b=*/false, b,
                /*c_mod=*/(short)0, c, /*reuse_a=*/false, /*reuse_b=*/false);
    }

    float* outp = pre + (size_t)row0 * F + col;
#pragma unroll
    for (int v = 0; v < 8; ++v)
        outp[(size_t)(v + (hi << 3)) * F] = c[v];
    (void)n_nodes;
}

// ---------------- edge scatter: out[dst] += val * pre[src] ----------------
// One wave (32 lanes) per edge; each lane handles 4 contiguous columns (float4).
__global__ __launch_bounds__(256) void scatter_add(const float* __restrict__ pre,
                                                   const float* __restrict__ vals,
                                                   const int* __restrict__ src,
                                                   const int* __restrict__ dst,
                                                   float* __restrict__ out,
                                                   int n_edges) {
    const int lane = threadIdx.x & 31;
    const long long e = (long long)blockIdx.x * 8 + (threadIdx.x >> 5);
    if (e >= n_edges) return;

    const int   s = src[e];
    const int   d = dst[e];
    const float v = vals[e];

    float4 m = ((const float4*)(pre + (size_t)s * F))[lane];   // 16B coalesced
    float* o = out + (size_t)d * F + lane * 4;
    atomicAdd(o + 0, v * m.x);   // -> global_atomic_add_f32
    atomicAdd(o + 1, v * m.y);
    atomicAdd(o + 2, v * m.z);
    atomicAdd(o + 3, v * m.w);
}

// ---------------- host-side launch ----------------

extern "C" void kernel_launch(void* const* d_in, const int* in_sizes, int n_in,
                              void* d_out, int out_size, void* d_ws, size_t ws_size,
                              hipStream_t stream) {
    const float* x  = (const float*)d_in[0];   // [N, 128]
    const float* Wb = (const float*)d_in[1];   // [4, 128, 128]
    const float* Wc = (const float*)d_in[2];   // [8, 4]
    const float* ev = (const float*)d_in[3];   // [8, E]
    const int*   es = (const int*)  d_in[4];   // [8, E]
    const int*   ed = (const int*)  d_in[5];   // [8, E]
    float* out = (float*)d_out;                // [N, 128]

    const int n_nodes = in_sizes[0] / F;       // 50000
    const int n_edges = in_sizes[4] / NREL;    // 500000

    // Workspace: W_rel (512 KB) + one per-relation pre buffer (25.6 MB)
    float* Wr  = (float*)d_ws;
    float* pre = Wr + (size_t)NREL * F * F;

    const long long out_n = (long long)n_nodes * F;
    zero_f32<<<1024, 256, 0, stream>>>(out, out_n);
    wrel_kernel<<<(NREL * F * F) / 256, 256, 0, stream>>>(Wb, Wc, Wr);

    const int gemm_blocks = n_nodes / 16;        // 3125 (N divisible by 16)
    const int sc_blocks   = (n_edges + 7) / 8;   // 8 edges per 256-thread block

    for (int r = 0; r < NREL; ++r) {
        gemm_wmma<<<gemm_blocks, 256, 0, stream>>>(x, Wr + (size_t)r * F * F, pre, n_nodes);
        scatter_add<<<sc_blocks, 256, 0, stream>>>(
            pre,
            ev + (size_t)r * n_edges,
            es + (size_t)r * n_edges,
            ed + (size_t)r * n_edges,
            out, n_edges);
    }

    relu_inplace<<<1024, 256, 0, stream>>>(out, out_n);
}